// KNN_73830487818478
// MI455X (gfx1250) — compile-verified
//
#include <hip/hip_runtime.h>

// KNN (K=16) + feature gather/concat for (8, 4096, 3) points, (8, 4096, 64) features.
// Distance ranking via V_WMMA_F32_16X16X4_F32:  score = |c|^2 - 2*q.c  (row-constant
// |q|^2 omitted -- does not change per-query ordering).

typedef float v2f __attribute__((ext_vector_type(2)));
typedef float v8f __attribute__((ext_vector_type(8)));

#define NPTS 4096
#define KNN  16
#define FEAT 64
#define WAVES_PER_BLOCK 4

__global__ __launch_bounds__(128) void KNN_73830487818478_kernel(
    const float* __restrict__ points,    // (8, 4096, 3)
    const float* __restrict__ features,  // (8, 4096, 64)
    float* __restrict__ out)             // (8, 4096, 16, 128)
{
    // score tile per wave: 32 query rows x 16 cols, stride 17 to avoid bank conflicts
    __shared__ float sc[WAVES_PER_BLOCK][32][17];
    __shared__ int   nbr[WAVES_PER_BLOCK][32][KNN];

    const int lane = threadIdx.x & 31;
    const int wv   = threadIdx.x >> 5;
    const int b    = blockIdx.x >> 5;     // 32 blocks per batch
    const int qblk = blockIdx.x & 31;
    const int qbase = qblk * 128 + wv * 32;   // this wave's 32 queries

    const int  m  = lane & 15;
    const bool hi = lane >= 16;

    const float* Pb = points + (size_t)b * NPTS * 3;

    // ---- A fragments: 16x4 f32 layout. VGPR0 = K0 (lanes 0-15) / K2 (lanes 16-31),
    //      VGPR1 = K1 / K3.  A row = (qx, qy, qz, 1).
    const int q0 = qbase + m;
    const int q1 = qbase + 16 + m;
    const float p0x = Pb[q0 * 3 + 0], p0y = Pb[q0 * 3 + 1], p0z = Pb[q0 * 3 + 2];
    const float p1x = Pb[q1 * 3 + 0], p1y = Pb[q1 * 3 + 1], p1z = Pb[q1 * 3 + 2];
    v2f a0, a1;
    a0.x = hi ? p0z : p0x;   a0.y = hi ? 1.0f : p0y;
    a1.x = hi ? p1z : p1x;   a1.y = hi ? 1.0f : p1y;

    const int qself = qbase + lane;   // query row owned by this lane in the scan phase

    // sorted descending: bestd[0] is the current worst (largest) kept score
    float bestd[KNN];
    int   bestidx[KNN];
#pragma unroll
    for (int j = 0; j < KNN; ++j) { bestd[j] = 3.0e38f; bestidx[j] = 0; }

    for (int ct = 0; ct < NPTS / 16; ++ct) {
        // ---- B fragment: 4x16, rows (-2cx, -2cy, -2cz, |c|^2).
        //      VGPR0 = K0 (lanes 0-15) / K2 (lanes 16-31), VGPR1 = K1 / K3; N = lane&15.
        const int cg = ct * 16 + m;
        const float cx = Pb[cg * 3 + 0], cy = Pb[cg * 3 + 1], cz = Pb[cg * 3 + 2];
        const float nrm = cx * cx + cy * cy + cz * cz;
        v2f bf;
        bf.x = hi ? (-2.0f * cz) : (-2.0f * cx);
        bf.y = hi ? nrm          : (-2.0f * cy);

        v8f c0 = {};
        v8f c1 = {};
        c0 = __builtin_amdgcn_wmma_f32_16x16x4_f32(false, a0, false, bf, (short)0, c0, false, false);
        c1 = __builtin_amdgcn_wmma_f32_16x16x4_f32(false, a1, false, bf, (short)0, c1, false, false);

        // C/D layout: VGPR v -> row v (lanes 0-15) or row v+8 (lanes 16-31), col = lane&15
        const int rbase = hi ? 8 : 0;
#pragma unroll
        for (int v = 0; v < 8; ++v) sc[wv][rbase + v][m] = c0[v];
#pragma unroll
        for (int v = 0; v < 8; ++v) sc[wv][16 + rbase + v][m] = c1[v];

        // same-wave LDS ops complete in order; only stop compiler reordering
        __builtin_amdgcn_wave_barrier();
        __asm__ __volatile__("" ::: "memory");

        // ---- per-lane top-K scan over this tile's 16 candidates (lane owns row `lane`)
#pragma unroll
        for (int t = 0; t < 16; ++t) {
            const float s  = sc[wv][lane][t];
            const int  col = ct * 16 + t;
            if (col != qself && s < bestd[0]) {
                bestd[0] = s; bestidx[0] = col;
#pragma unroll
                for (int j = 0; j < KNN - 1; ++j) {
                    if (bestd[j] < bestd[j + 1]) {
                        const float td = bestd[j]; bestd[j] = bestd[j + 1]; bestd[j + 1] = td;
                        const int   ti = bestidx[j]; bestidx[j] = bestidx[j + 1]; bestidx[j + 1] = ti;
                    }
                }
            }
        }
        __builtin_amdgcn_wave_barrier();
        __asm__ __volatile__("" ::: "memory");
    }

    // ---- publish neighbor lists nearest-first (ascending distance = reversed list)
#pragma unroll
    for (int k = 0; k < KNN; ++k) nbr[wv][lane][k] = bestidx[KNN - 1 - k];
    __builtin_amdgcn_wave_barrier();
    __asm__ __volatile__("" ::: "memory");

    // ---- fused gather + concat, float4 coalesced:
    //      lanes 0-15 write center feats [0..63], lanes 16-31 write neighbor feats [64..127]
    const float4* Fb4 = (const float4*)(features + (size_t)b * NPTS * FEAT);
    float4* out4 = (float4*)out;
    for (int qq = 0; qq < 32; ++qq) {
        const int q = qbase + qq;
        for (int k = 0; k < KNN; ++k) {
            const int nb   = nbr[wv][qq][k];
            const int srcp = hi ? nb : q;
            const float4 v4 = Fb4[srcp * (FEAT / 4) + m];
            const size_t orow = (((size_t)b * NPTS + q) * KNN + k) * 32; // 128 floats = 32 float4
            out4[orow + lane] = v4;
        }
    }
}

extern "C" void kernel_launch(void* const* d_in, const int* in_sizes, int n_in,
                              void* d_out, int out_size, void* d_ws, size_t ws_size,
                              hipStream_t stream) {
    (void)in_sizes; (void)n_in; (void)d_ws; (void)ws_size; (void)out_size;
    const float* points   = (const float*)d_in[0];   // (8, 4096, 3)  f32
    const float* features = (const float*)d_in[1];   // (8, 4096, 64) f32
    float* out = (float*)d_out;                      // (8, 4096, 16, 128) f32

    // 8 batches x 32 query-blocks of 128 queries; 128 threads = 4 waves,
    // each wave owns 32 queries (two 16-row WMMA A tiles).
    dim3 grid(8 * 32);
    dim3 block(128);
    KNN_73830487818478_kernel<<<grid, block, 0, stream>>>(points, features, out);
}